// PythonNTMCell_54906861912660
// MI455X (gfx1250) — compile-verified
//
#include <hip/hip_runtime.h>

typedef float v2f __attribute__((ext_vector_type(2)));
typedef float v8f __attribute__((ext_vector_type(8)));

#define TT 512
#define BB 32
#define DD 1024
#define NN 128
#define TB (TT * BB)   // 16384 rows for the projection GEMM

// ---------------------------------------------------------------------------
// Phase 1: fused 5-way projection GEMM with V_WMMA_F32_16X16X4_F32.
// Grid: TB/16 = 1024 blocks of 256 threads (8 waves).
// Each block handles one 16-row M tile; wave w owns N columns [w*16, w*16+16)
// across all 5 weight matrices (5 accumulator tiles per wave).
// ---------------------------------------------------------------------------
__global__ __launch_bounds__(256) void ntm_proj_wmma(
    const float* __restrict__ x,
    const float* __restrict__ Wk, const float* __restrict__ Wv,
    const float* __restrict__ Wq, const float* __restrict__ We,
    const float* __restrict__ Ww,
    float* __restrict__ Pk, float* __restrict__ Pv,
    float* __restrict__ Pq, float* __restrict__ Pe,
    float* __restrict__ Pw)
{
    const int lane  = threadIdx.x & 31;
    const int wave  = threadIdx.x >> 5;
    const int mbase = blockIdx.x * 16;
    const int m     = lane & 15;          // M index within tile for A; N offset for B/D
    const int kb    = (lane >> 4) * 2;    // K sub-offset for this lane half
    const int ncol  = wave * 16;          // N tile base for this wave
    const int n     = ncol + m;           // N column this lane touches in B/D

    const float* W[5] = {Wk, Wv, Wq, We, Ww};
    float*       P[5] = {Pk, Pv, Pq, Pe, Pw};

    v8f acc[5] = {};

    const float* arow = x + (size_t)(mbase + m) * DD + kb;
    const float* brow[5];
#pragma unroll
    for (int j = 0; j < 5; ++j) brow[j] = W[j] + (size_t)n * DD + kb;

    for (int k0 = 0; k0 < DD; k0 += 4) {
        v2f a = *(const v2f*)(arow + k0);       // A[m][k0+kb .. +1]
#pragma unroll
        for (int j = 0; j < 5; ++j) {
            v2f b = *(const v2f*)(brow[j] + k0); // B[k0+kb..+1][n] = W[n][k..]
            acc[j] = __builtin_amdgcn_wmma_f32_16x16x4_f32(
                false, a, false, b, (short)0, acc[j], false, false);
        }
    }

    // D tile: VGPR r -> element [M = r + 8*(lane>>4)][N = lane&15]
    const int row0 = mbase + 8 * (lane >> 4);
#pragma unroll
    for (int j = 0; j < 5; ++j) {
        float* p = P[j];
#pragma unroll
        for (int r = 0; r < 8; ++r)
            p[(size_t)(row0 + r) * NN + n] = acc[j][r];
    }
}

// ---------------------------------------------------------------------------
// fast transcendentals via native TRANS units (v_exp_f32 / v_rcp_f32)
// ---------------------------------------------------------------------------
__device__ __forceinline__ float fast_tanh(float x) {
    // tanh(x) = 1 - 2/(e^{2x}+1); saturates correctly at +/-inf via rcp.
    float e = __builtin_amdgcn_exp2f(x * 2.885390081777927f); // 2*log2(e)
    return 1.0f - 2.0f * __builtin_amdgcn_rcpf(e + 1.0f);
}
__device__ __forceinline__ float fast_sigmoid(float x) {
    float e = __builtin_amdgcn_exp2f(-x * 1.4426950408889634f);
    return __builtin_amdgcn_rcpf(1.0f + e);
}

// ---------------------------------------------------------------------------
// Phase 2: recurrent scan. Grid: B*8 = 256 workgroups of 256 threads.
// Block (b, slice) owns rows [slice*16, slice*16+16) of S_b, held entirely in
// registers (8 floats per thread). Per step: stream k/q (128 each) and
// e/w/v (16 each) for this slice into LDS, in-wave shfl reductions for the
// k-norm and the Sq row sums.
// ---------------------------------------------------------------------------
__global__ __launch_bounds__(256) void ntm_scan(
    const float* __restrict__ S0,
    const float* __restrict__ Pk, const float* __restrict__ Pv,
    const float* __restrict__ Pq, const float* __restrict__ Pe,
    const float* __restrict__ Pw,
    float* __restrict__ out,    // (T,B,N)
    float* __restrict__ outS)   // (B,N,N)
{
    __shared__ float kbuf[NN];
    __shared__ float qbuf[NN];
    __shared__ float ebuf[16];
    __shared__ float wbuf[16];
    __shared__ float vbuf[16];

    const int tid  = threadIdx.x;
    const int lane = tid & 31;
    const int b    = blockIdx.x >> 3;
    const int r0   = (blockIdx.x & 7) * 16;
    const int il   = tid >> 4;        // local row 0..15
    const int jg   = (tid & 15) * 8;  // column base 0..120

    // S slice lives in registers for the whole scan
    const float* s0p = S0 + (size_t)b * NN * NN + (size_t)(r0 + il) * NN + jg;
    float4 sa = *(const float4*)(s0p);
    float4 sb = *(const float4*)(s0p + 4);

    for (int t = 0; t < TT; ++t) {
        const int base = (t * BB + b) * NN;

        __syncthreads();   // previous step done reading the staging buffers
        if (tid < 128)       kbuf[tid]       = Pk[base + tid];
        else                 qbuf[tid - 128] = Pq[base + tid - 128];
        if (tid < 16)        ebuf[tid]       = Pe[base + r0 + tid];
        else if (tid < 32)   wbuf[tid - 16]  = Pw[base + r0 + tid - 16];
        else if (tid < 48)   vbuf[tid - 32]  = Pv[base + r0 + tid - 32];
        __syncthreads();

        // ||k||: each wave reduces all 128 squares redundantly (no barrier)
        float ss = 0.0f;
#pragma unroll
        for (int i = 0; i < 4; ++i) {
            float kv = kbuf[lane + 32 * i];
            ss += kv * kv;
        }
#pragma unroll
        for (int msk = 16; msk >= 1; msk >>= 1)
            ss += __shfl_xor(ss, msk, 32);
        const float inv = __builtin_amdgcn_rcpf(__builtin_sqrtf(ss) + 1e-6f);

        const float e_i = ebuf[il];
        const float wv  = wbuf[il] * vbuf[il];

        float accq = 0.0f;
#pragma unroll
        for (int c = 0; c < 4; ++c) {
            float kn = kbuf[jg + c] * inv;
            float s  = (&sa.x)[c];
            s = fast_tanh(s * (1.0f - e_i * kn) + wv * kn);
            (&sa.x)[c] = s;
            accq += s * qbuf[jg + c];
        }
#pragma unroll
        for (int c = 0; c < 4; ++c) {
            float kn = kbuf[jg + 4 + c] * inv;
            float s  = (&sb.x)[c];
            s = fast_tanh(s * (1.0f - e_i * kn) + wv * kn);
            (&sb.x)[c] = s;
            accq += s * qbuf[jg + 4 + c];
        }

        // Sq[i]: reduce across the 16 threads of this row (masks stay in-half)
#pragma unroll
        for (int msk = 8; msk >= 1; msk >>= 1)
            accq += __shfl_xor(accq, msk, 32);

        if ((tid & 15) == 0) {
            float y = accq * accq * fast_sigmoid(accq);  // Sq * silu(Sq)
            out[base + r0 + il] = y;
        }
    }

    // write final state slice
    float* sp = outS + (size_t)b * NN * NN + (size_t)(r0 + il) * NN + jg;
    *(float4*)(sp)     = sa;
    *(float4*)(sp + 4) = sb;
}

// ---------------------------------------------------------------------------
extern "C" void kernel_launch(void* const* d_in, const int* in_sizes, int n_in,
                              void* d_out, int out_size, void* d_ws, size_t ws_size,
                              hipStream_t stream) {
    const float* x  = (const float*)d_in[0];
    const float* S0 = (const float*)d_in[1];
    const float* Wk = (const float*)d_in[2];
    const float* Wv = (const float*)d_in[3];
    const float* Wq = (const float*)d_in[4];
    const float* We = (const float*)d_in[5];
    const float* Ww = (const float*)d_in[6];
    float* out = (float*)d_out;

    float* ws = (float*)d_ws;
    const size_t PSZ = (size_t)TB * NN;   // 2,097,152 floats per projection
    float* Pk = ws + 0 * PSZ;
    float* Pv = ws + 1 * PSZ;
    float* Pq = ws + 2 * PSZ;
    float* Pe = ws + 3 * PSZ;
    float* Pw = ws + 4 * PSZ;

    ntm_proj_wmma<<<TB / 16, 256, 0, stream>>>(x, Wk, Wv, Wq, We, Ww,
                                               Pk, Pv, Pq, Pe, Pw);
    ntm_scan<<<BB * 8, 256, 0, stream>>>(S0, Pk, Pv, Pq, Pe, Pw,
                                         out, out + PSZ);
}